// PhiMoE_4990751998306
// MI455X (gfx1250) — compile-verified
//
#include <hip/hip_runtime.h>
#include <hip/hip_bf16.h>

// ---------------- problem constants ----------------
constexpr int   TOK  = 1024;   // tokens
constexpr int   HID  = 2048;   // hidden
constexpr int   IMID = 3200;   // intermediate
constexpr int   NE   = 8;      // experts
constexpr float JEPS = 0.01f;  // jitter eps

constexpr int MT     = 128;                // M tile rows per block (8 WMMA sub-tiles)
constexpr int MAXA   = 2 * TOK + NE * MT;  // padded assignment capacity = 3072
constexpr int NBLK   = MAXA / MT;          // 24 M-blocks
constexpr int KT     = 64;                 // K staging tile
constexpr int LDST   = 68;                 // LDS row stride (floats): 16B-aligned, conflict-free

typedef float v2f __attribute__((ext_vector_type(2)));
typedef float v4f __attribute__((ext_vector_type(4)));
typedef float v8f __attribute__((ext_vector_type(8)));

// -------- async global->LDS (gfx1250) with VGPR-staging fallback --------
#if defined(__has_builtin)
#if __has_builtin(__builtin_amdgcn_global_load_async_to_lds_b128) && \
    __has_builtin(__builtin_amdgcn_s_wait_asynccnt)
#define HAS_ASYNC 1
#endif
#endif
#ifndef HAS_ASYNC
#define HAS_ASYNC 0
#endif

typedef int v4i_vec __attribute__((vector_size(16)));
typedef __attribute__((address_space(1))) v4i_vec g_v4i;  // global
typedef __attribute__((address_space(3))) v4i_vec l_v4i;  // LDS

__device__ __forceinline__ void copy16(const float* g, float* l) {
#if HAS_ASYNC
  __builtin_amdgcn_global_load_async_to_lds_b128(
      (g_v4i*)g, (l_v4i*)l, 0, 0);
#else
  v4f v = *(const v4f*)g;
  l[0] = v[0]; l[1] = v[1]; l[2] = v[2]; l[3] = v[3];
#endif
}

__device__ __forceinline__ void async_wait() {
#if HAS_ASYNC
  __builtin_amdgcn_s_wait_asynccnt(0);
#endif
}

// ---------------- kernel 0: zero out + counters ----------------
__global__ void zero_kernel(float* __restrict__ out, int n,
                            int* __restrict__ counts, int* __restrict__ cursor) {
  int i = blockIdx.x * blockDim.x + threadIdx.x;
  for (; i < n; i += gridDim.x * blockDim.x) out[i] = 0.0f;
  if (blockIdx.x == 0 && threadIdx.x < NE) {
    counts[threadIdx.x] = 0;
    cursor[threadIdx.x] = 0;
  }
}

// ---------------- kernel 1: router logits + sparsemixer ----------------
__global__ __launch_bounds__(256)
void router_kernel(const float* __restrict__ x, const float* __restrict__ gw,
                   float* __restrict__ topw, int* __restrict__ topi,
                   int* __restrict__ counts) {
  const int t = blockIdx.x;
  __shared__ float logits[NE];

  const int wave = threadIdx.x >> 5;  // 8 waves -> 8 experts
  const int lane = threadIdx.x & 31;
  const float* xr = x + (size_t)t * HID;
  const float* g  = gw + (size_t)wave * HID;

  float s = 0.0f;
  for (int h = lane; h < HID; h += 32) s += xr[h] * g[h];
  #pragma unroll
  for (int off = 16; off > 0; off >>= 1) s += __shfl_xor(s, off, 32);
  if (lane == 0) logits[wave] = s;
  __syncthreads();

  if (threadIdx.x == 0) {
    const float INF = __builtin_inff();
    float sc[NE];
    #pragma unroll
    for (int e = 0; e < NE; ++e) sc[e] = logits[e];

    // ---- top-1 (first-max argmax) ----
    float max1 = sc[0]; int ind1 = 0;
    #pragma unroll
    for (int e = 1; e < NE; ++e) if (sc[e] > max1) { max1 = sc[e]; ind1 = e; }

    float g1[NE]; float mx1 = -INF;
    #pragma unroll
    for (int e = 0; e < NE; ++e) {
      float factor = fmaxf(fabsf(sc[e]), max1);
      bool mask = (max1 - sc[e]) / factor > 2.0f * JEPS;
      g1[e] = mask ? -INF : sc[e];
      mx1 = fmaxf(mx1, g1[e]);
    }
    float sum1 = 0.0f;
    #pragma unroll
    for (int e = 0; e < NE; ++e) { g1[e] = __expf(g1[e] - mx1); sum1 += g1[e]; }
    float mult1 = g1[ind1] / sum1;

    // ---- top-2: mask out ind1, redo ----
    float ms[NE];
    #pragma unroll
    for (int e = 0; e < NE; ++e) ms[e] = (e == ind1) ? -INF : sc[e];
    float max2 = ms[0]; int ind2 = 0;
    #pragma unroll
    for (int e = 1; e < NE; ++e) if (ms[e] > max2) { max2 = ms[e]; ind2 = e; }

    float g2[NE]; float mx2 = -INF;
    #pragma unroll
    for (int e = 0; e < NE; ++e) {
      float factor = fmaxf(fabsf(sc[e]), max2);
      bool mask = (max2 - sc[e]) / factor > 2.0f * JEPS;
      g2[e] = (mask || e == ind1) ? -INF : sc[e];
      mx2 = fmaxf(mx2, g2[e]);
    }
    float sum2 = 0.0f;
    #pragma unroll
    for (int e = 0; e < NE; ++e) { g2[e] = __expf(g2[e] - mx2); sum2 += g2[e]; }
    float mult2 = g2[ind2] / sum2;

    topw[t * 2 + 0] = mult1;  topw[t * 2 + 1] = mult2;
    topi[t * 2 + 0] = ind1;   topi[t * 2 + 1] = ind2;
    atomicAdd(&counts[ind1], 1);
    atomicAdd(&counts[ind2], 1);
  }
}

// ---------------- kernel 2: segment offsets + scatter assignments ----------------
__global__ __launch_bounds__(256)
void scan_scatter_kernel(const float* __restrict__ topw, const int* __restrict__ topi,
                         const int* __restrict__ counts, int* __restrict__ cursor,
                         int* __restrict__ tileExpert,
                         int* __restrict__ assignTok, float* __restrict__ assignGate) {
  __shared__ int soff[NE + 1];
  if (threadIdx.x == 0) {
    int off = 0;
    for (int e = 0; e < NE; ++e) {
      soff[e] = off;
      off += (counts[e] + (MT - 1)) & ~(MT - 1);  // pad segments to MT rows
    }
    soff[NE] = off;
    for (int r = 0; r < NBLK; ++r) {
      int base = r * MT;
      int ex = -1;
      for (int e = 0; e < NE; ++e)
        if (base >= soff[e] && base < soff[e + 1]) ex = e;
      tileExpert[r] = ex;
    }
  }
  __syncthreads();
  // default-fill (padded rows: token 0, gate 0 -> exact-zero contribution)
  for (int a = threadIdx.x; a < MAXA; a += blockDim.x) {
    assignTok[a] = 0; assignGate[a] = 0.0f;
  }
  __syncthreads();
  for (int idx = threadIdx.x; idx < 2 * TOK; idx += blockDim.x) {
    int t = idx >> 1, s = idx & 1;
    int e = topi[t * 2 + s];
    int pos = soff[e] + atomicAdd(&cursor[e], 1);
    assignTok[pos]  = t;
    assignGate[pos] = topw[t * 2 + s];
  }
}

// ---------------- kernel 3: up-proj (h1, h3) + SiLU gate -> act ----------------
__global__ __launch_bounds__(128)
void up_proj_kernel(const float* __restrict__ x,
                    const float* __restrict__ w1, const float* __restrict__ w3,
                    const int* __restrict__ assignTok, const int* __restrict__ tileExpert,
                    float* __restrict__ act) {
  const int r = blockIdx.x;
  const int e = tileExpert[r];
  if (e < 0) return;
  const int i0 = blockIdx.y * 64;

  __shared__ float xs [MT * LDST];
  __shared__ float w1s[64 * LDST];
  __shared__ float w3s[64 * LDST];
  __shared__ int   tok[MT];

  tok[threadIdx.x] = assignTok[r * MT + threadIdx.x];   // 128 threads, MT=128

  const float* w1e = w1 + (size_t)e * IMID * HID;
  const float* w3e = w3 + (size_t)e * IMID * HID;

  const int lane = threadIdx.x & 31;
  const int wave = threadIdx.x >> 5;
  const int mn   = lane & 15;       // A row (within sub-tile) / B col n
  const int half = lane >> 4;       // K-half select

  v8f acc1[8]; v8f acc3[8];
  #pragma unroll
  for (int s = 0; s < 8; ++s) { acc1[s] = (v8f){}; acc3[s] = (v8f){}; }

  for (int k0 = 0; k0 < HID; k0 += KT) {
    __syncthreads();
    // stage x tile: 128 rows x 16 float4
    #pragma unroll
    for (int j = 0; j < 16; ++j) {
      int f = threadIdx.x + j * 128;            // 0..2047
      int row = f >> 4, c4 = f & 15;
      copy16(x + (size_t)tok[row] * HID + k0 + c4 * 4, &xs[row * LDST + c4 * 4]);
    }
    // stage w1/w3 tiles: 64 rows x 16 float4 each (L2 catches cross-block reuse)
    #pragma unroll
    for (int j = 0; j < 8; ++j) {
      int f = threadIdx.x + j * 128;            // 0..1023
      int row = f >> 4, c4 = f & 15;
      size_t goff = (size_t)(i0 + row) * HID + k0 + c4 * 4;
      copy16(w1e + goff, &w1s[row * LDST + c4 * 4]);
      copy16(w3e + goff, &w3s[row * LDST + c4 * 4]);
    }
    async_wait();
    __syncthreads();

    for (int kk = 0; kk < KT; kk += 4) {
      const int kc = kk + 2 * half;
      const int br = (wave * 16 + mn) * LDST;
      v2f b1; b1[0] = w1s[br + kc]; b1[1] = w1s[br + kc + 1];
      v2f b3; b3[0] = w3s[br + kc]; b3[1] = w3s[br + kc + 1];
      #pragma unroll
      for (int s = 0; s < 8; ++s) {
        const int ar = (s * 16 + mn) * LDST;
        v2f a; a[0] = xs[ar + kc]; a[1] = xs[ar + kc + 1];
        acc1[s] = __builtin_amdgcn_wmma_f32_16x16x4_f32(false, a, false, b1, (short)0, acc1[s], false, false);
        acc3[s] = __builtin_amdgcn_wmma_f32_16x16x4_f32(false, a, false, b3, (short)0, acc3[s], false, false);
      }
    }
  }

  // epilogue: act = silu(h1) * h3
  const int col = i0 + wave * 16 + mn;
  #pragma unroll
  for (int s = 0; s < 8; ++s) {
    #pragma unroll
    for (int v = 0; v < 8; ++v) {
      int m = s * 16 + v + 8 * half;
      float h1 = acc1[s][v], h3 = acc3[s][v];
      float sil = h1 / (1.0f + __expf(-h1));
      act[(size_t)(r * MT + m) * IMID + col] = sil * h3;
    }
  }
}

// ---------------- kernel 4: down-proj + gate + combine ----------------
__global__ __launch_bounds__(128)
void down_proj_kernel(const float* __restrict__ act, const float* __restrict__ w2,
                      const int* __restrict__ assignTok, const float* __restrict__ assignGate,
                      const int* __restrict__ tileExpert, float* __restrict__ out) {
  const int r = blockIdx.x;
  const int e = tileExpert[r];
  if (e < 0) return;
  const int h0 = blockIdx.y * 64;

  __shared__ float as  [MT * LDST];
  __shared__ float wsld[64 * LDST];
  __shared__ int   tok [MT];
  __shared__ float gate[MT];

  tok [threadIdx.x] = assignTok [r * MT + threadIdx.x];
  gate[threadIdx.x] = assignGate[r * MT + threadIdx.x];

  const float* w2e = w2 + (size_t)e * HID * IMID;

  const int lane = threadIdx.x & 31;
  const int wave = threadIdx.x >> 5;
  const int mn   = lane & 15;
  const int half = lane >> 4;

  v8f acc[8];
  #pragma unroll
  for (int s = 0; s < 8; ++s) acc[s] = (v8f){};

  for (int k0 = 0; k0 < IMID; k0 += KT) {
    __syncthreads();
    #pragma unroll
    for (int j = 0; j < 16; ++j) {
      int f = threadIdx.x + j * 128;
      int row = f >> 4, c4 = f & 15;
      copy16(act + (size_t)(r * MT + row) * IMID + k0 + c4 * 4, &as[row * LDST + c4 * 4]);
    }
    #pragma unroll
    for (int j = 0; j < 8; ++j) {
      int f = threadIdx.x + j * 128;
      int row = f >> 4, c4 = f & 15;
      copy16(w2e + (size_t)(h0 + row) * IMID + k0 + c4 * 4, &wsld[row * LDST + c4 * 4]);
    }
    async_wait();
    __syncthreads();

    for (int kk = 0; kk < KT; kk += 4) {
      const int kc = kk + 2 * half;
      const int br = (wave * 16 + mn) * LDST;
      v2f b; b[0] = wsld[br + kc]; b[1] = wsld[br + kc + 1];
      #pragma unroll
      for (int s = 0; s < 8; ++s) {
        const int ar = (s * 16 + mn) * LDST;
        v2f a; a[0] = as[ar + kc]; a[1] = as[ar + kc + 1];
        acc[s] = __builtin_amdgcn_wmma_f32_16x16x4_f32(false, a, false, b, (short)0, acc[s], false, false);
      }
    }
  }

  const int col = h0 + wave * 16 + mn;
  #pragma unroll
  for (int s = 0; s < 8; ++s) {
    #pragma unroll
    for (int v = 0; v < 8; ++v) {
      int m = s * 16 + v + 8 * half;
      atomicAdd(&out[(size_t)tok[m] * HID + col], acc[s][v] * gate[m]);
    }
  }
}

// ---------------- launcher ----------------
extern "C" void kernel_launch(void* const* d_in, const int* in_sizes, int n_in,
                              void* d_out, int out_size, void* d_ws, size_t ws_size,
                              hipStream_t stream) {
  const float* x   = (const float*)d_in[0];  // [T, H]
  const float* gw  = (const float*)d_in[1];  // [E, H]
  const float* w1  = (const float*)d_in[2];  // [E, I, H]
  const float* w3  = (const float*)d_in[3];  // [E, I, H]
  const float* w2  = (const float*)d_in[4];  // [E, H, I]
  float* out = (float*)d_out;                // [T, H]

  char* ws = (char*)d_ws;
  size_t off = 0;
  auto take = [&](size_t bytes) -> char* {
    char* p = ws + off;
    off = (off + bytes + 255) & ~(size_t)255;
    return p;
  };
  float* topw       = (float*)take((size_t)TOK * 2 * sizeof(float));
  int*   topi       = (int*)  take((size_t)TOK * 2 * sizeof(int));
  int*   counts     = (int*)  take(NE * sizeof(int));
  int*   cursor     = (int*)  take(NE * sizeof(int));
  int*   tileExpert = (int*)  take(NBLK * sizeof(int));
  int*   assignTok  = (int*)  take(MAXA * sizeof(int));
  float* assignGate = (float*)take(MAXA * sizeof(float));
  float* act        = (float*)take((size_t)MAXA * IMID * sizeof(float));  // ~39.3 MB

  zero_kernel<<<512, 256, 0, stream>>>(out, TOK * HID, counts, cursor);
  router_kernel<<<TOK, 256, 0, stream>>>(x, gw, topw, topi, counts);
  scan_scatter_kernel<<<1, 256, 0, stream>>>(topw, topi, counts, cursor,
                                             tileExpert, assignTok, assignGate);
  up_proj_kernel<<<dim3(NBLK, IMID / 64), 128, 0, stream>>>(
      x, w1, w3, assignTok, tileExpert, act);
  down_proj_kernel<<<dim3(NBLK, HID / 64), 128, 0, stream>>>(
      act, w2, assignTok, assignGate, tileExpert, out);
}